// MaxKGraphConv_51161650430038
// MI455X (gfx1250) — compile-verified
//
#include <hip/hip_runtime.h>
#include <hip/hip_bf16.h>

typedef __attribute__((ext_vector_type(2))) float v2f;
typedef __attribute__((ext_vector_type(8))) float v8f;

#define IN_FEATS  256
#define OUT_FEATS 64
#define W_PAIRS   ((IN_FEATS / 2) * OUT_FEATS)   // 8192 float2 = 64 KB LDS

// ---------------------------------------------------------------------------
// Zero a float region
// ---------------------------------------------------------------------------
__global__ void gcn_zero_kernel(float* __restrict__ p, int n) {
    int t = blockIdx.x * blockDim.x + threadIdx.x;
    if (t < n) p[t] = 0.0f;
}

// ---------------------------------------------------------------------------
// Degree counting: out_cnt[src[e]] += 1 ; in_cnt[dst[e]] += 1
// ---------------------------------------------------------------------------
__global__ void gcn_degree_kernel(const int* __restrict__ src,
                                  const int* __restrict__ dst,
                                  float* __restrict__ out_cnt,
                                  float* __restrict__ in_cnt,
                                  int n_edges) {
    int e = blockIdx.x * blockDim.x + threadIdx.x;
    if (e >= n_edges) return;
    atomicAdd(&out_cnt[src[e]], 1.0f);
    atomicAdd(&in_cnt[dst[e]], 1.0f);
}

// ---------------------------------------------------------------------------
// h = (feat * outdeg^-1/2) @ weight, via V_WMMA_F32_16X16X4_F32.
//
// Weight is staged in LDS *pre-paired in WMMA B-operand layout*:
//   wlds[k/2 * 64 + n] = { w[k][n], w[k+1][n] }   (k even)
// so each B operand is a single contiguous ds_load_b64 into the even-aligned
// VGPR pair — no repacking movs. Each wave32 computes TWO 16-row M-tiles
// (32x64 output, 8 v8f accumulators); per 4-k step:
//   2 global b64 A-loads + 2 v_pk_mul (outdeg scale) + 4 ds_load_b64 + 8 WMMAs.
//
// A 16x4 f32 layout: lanes 0-15 -> M=lane, K={k,k+1}; lanes 16-31 -> K={k+2,k+3}.
// B 4x16 mirrors A's K split across lane halves; N = lane&15.
// C/D 16x16: VGPR v -> M=v (lanes 0-15) / M=v+8 (lanes 16-31), N = lane&15.
// ---------------------------------------------------------------------------
__global__ void __launch_bounds__(256)
gcn_gemm_wmma_kernel(const float* __restrict__ feat,
                     const float* __restrict__ weight,
                     const float* __restrict__ out_cnt,
                     float* __restrict__ h,
                     int n_tiles) {
    __shared__ v2f wlds[W_PAIRS];

    // Stage weight -> LDS in paired-K layout (one-time gather, 64 reuses)
    for (int i = threadIdx.x; i < W_PAIRS; i += blockDim.x) {
        const int k2 = i >> 6;          // pair index (k = 2*k2)
        const int n  = i & 63;
        v2f w;
        w.x = weight[(size_t)(2 * k2) * OUT_FEATS + n];
        w.y = weight[(size_t)(2 * k2 + 1) * OUT_FEATS + n];
        wlds[i] = w;
    }
    __syncthreads();

    const int lane  = threadIdx.x & 31;
    const int wave  = threadIdx.x >> 5;
    const int tile0 = (blockIdx.x * (blockDim.x >> 5) + wave) * 2;
    if (tile0 >= n_tiles) return;            // wave-uniform: EXEC stays all-1s
    const bool has1 = (tile0 + 1) < n_tiles; // wave-uniform

    const int mr   = lane & 15;
    const int kh   = lane >> 4;              // lane-half (K offset = 2*kh)
    const int ncol = lane & 15;              // B/D column

    const int row0 = tile0 * 16 + mr;
    const int row1 = has1 ? (tile0 + 1) * 16 + mr : row0;  // clamp tail loads

    const float s0 = rsqrtf(fmaxf(out_cnt[row0], 1.0f));
    const float s1 = rsqrtf(fmaxf(out_cnt[row1], 1.0f));
    const float* __restrict__ a0p = feat + (size_t)row0 * IN_FEATS + 2 * kh;
    const float* __restrict__ a1p = feat + (size_t)row1 * IN_FEATS + 2 * kh;

    v8f c00 = {}, c01 = {}, c02 = {}, c03 = {};
    v8f c10 = {}, c11 = {}, c12 = {}, c13 = {};

    for (int k = 0; k < IN_FEATS; k += 4) {
        v2f a0, a1;
        a0.x = a0p[k] * s0;  a0.y = a0p[k + 1] * s0;
        a1.x = a1p[k] * s1;  a1.y = a1p[k + 1] * s1;

        // B operands: contiguous float2 in LDS, immediate offsets 0/16/32/48
        const v2f* __restrict__ bp = wlds + ((k >> 1) + kh) * OUT_FEATS + ncol;
        const v2f b0 = bp[0];
        const v2f b1 = bp[16];
        const v2f b2 = bp[32];
        const v2f b3 = bp[48];

        c00 = __builtin_amdgcn_wmma_f32_16x16x4_f32(false, a0, false, b0, (short)0, c00, false, false);
        c10 = __builtin_amdgcn_wmma_f32_16x16x4_f32(false, a1, false, b0, (short)0, c10, false, false);
        c01 = __builtin_amdgcn_wmma_f32_16x16x4_f32(false, a0, false, b1, (short)0, c01, false, false);
        c11 = __builtin_amdgcn_wmma_f32_16x16x4_f32(false, a1, false, b1, (short)0, c11, false, false);
        c02 = __builtin_amdgcn_wmma_f32_16x16x4_f32(false, a0, false, b2, (short)0, c02, false, false);
        c12 = __builtin_amdgcn_wmma_f32_16x16x4_f32(false, a1, false, b2, (short)0, c12, false, false);
        c03 = __builtin_amdgcn_wmma_f32_16x16x4_f32(false, a0, false, b3, (short)0, c03, false, false);
        c13 = __builtin_amdgcn_wmma_f32_16x16x4_f32(false, a1, false, b3, (short)0, c13, false, false);
    }

    const int rowoff = kh * 8;
#pragma unroll
    for (int v = 0; v < 8; ++v) {
        const size_t r0 = (size_t)(tile0 * 16 + rowoff + v) * OUT_FEATS + ncol;
        h[r0 +  0] = c00[v];
        h[r0 + 16] = c01[v];
        h[r0 + 32] = c02[v];
        h[r0 + 48] = c03[v];
    }
    if (has1) {
#pragma unroll
        for (int v = 0; v < 8; ++v) {
            const size_t r1 = (size_t)((tile0 + 1) * 16 + rowoff + v) * OUT_FEATS + ncol;
            h[r1 +  0] = c10[v];
            h[r1 + 16] = c11[v];
            h[r1 + 32] = c12[v];
            h[r1 + 48] = c13[v];
        }
    }
}

// ---------------------------------------------------------------------------
// Edge scatter-add: accum[dst[e], :] += h[src[e], :]
// 16 threads per edge, float4 gather + 4 global atomic f32 adds (L2-resident).
// ---------------------------------------------------------------------------
__global__ void gcn_scatter_kernel(const float* __restrict__ h,
                                   const int* __restrict__ src,
                                   const int* __restrict__ dst,
                                   float* __restrict__ accum,
                                   int n_edges) {
    int t = blockIdx.x * blockDim.x + threadIdx.x;
    int e = t >> 4;
    if (e >= n_edges) return;
    int f = (t & 15) * 4;
    int sn = src[e], dn = dst[e];
    const float4 v = *(const float4*)(h + (size_t)sn * OUT_FEATS + f);
    float* o = accum + (size_t)dn * OUT_FEATS + f;
    atomicAdd(o + 0, v.x);
    atomicAdd(o + 1, v.y);
    atomicAdd(o + 2, v.z);
    atomicAdd(o + 3, v.w);
}

// ---------------------------------------------------------------------------
// Finalize in place: out = out * indeg^-1/2 + bias
// ---------------------------------------------------------------------------
__global__ void gcn_finalize_kernel(float* __restrict__ out,
                                    const float* __restrict__ in_cnt,
                                    const float* __restrict__ bias,
                                    int total) {
    int t = blockIdx.x * blockDim.x + threadIdx.x;
    if (t >= total) return;
    int node = t >> 6;           // /OUT_FEATS
    int f    = t & 63;
    float s = rsqrtf(fmaxf(in_cnt[node], 1.0f));
    out[t] = out[t] * s + bias[f];
}

// ---------------------------------------------------------------------------
extern "C" void kernel_launch(void* const* d_in, const int* in_sizes, int n_in,
                              void* d_out, int out_size, void* d_ws, size_t ws_size,
                              hipStream_t stream) {
    const float* feat   = (const float*)d_in[0];   // [N, 256]
    const float* weight = (const float*)d_in[1];   // [256, 64]
    const float* bias   = (const float*)d_in[2];   // [64]
    const int*   src    = (const int*)d_in[3];     // [E]
    const int*   dst    = (const int*)d_in[4];     // [E]
    float*       out    = (float*)d_out;           // [N, 64]

    const int n_nodes = in_sizes[0] / IN_FEATS;
    const int n_edges = in_sizes[3];

    // Workspace layout: [out_cnt N][in_cnt N][h N*64]
    float* out_cnt = (float*)d_ws;
    float* in_cnt  = out_cnt + n_nodes;
    float* h       = in_cnt + n_nodes;

    const int B = 256;

    // 1) zero counters + output accumulator (d_out is poisoned by harness)
    {
        int n = 2 * n_nodes;
        gcn_zero_kernel<<<(n + B - 1) / B, B, 0, stream>>>(out_cnt, n);
        int m = n_nodes * OUT_FEATS;
        gcn_zero_kernel<<<(m + B - 1) / B, B, 0, stream>>>(out, m);
    }

    // 2) degree counts
    gcn_degree_kernel<<<(n_edges + B - 1) / B, B, 0, stream>>>(src, dst, out_cnt, in_cnt, n_edges);

    // 3) WMMA GEMM with fused out-degree normalization (2 M-tiles per wave)
    {
        int n_tiles = (n_nodes + 15) / 16;               // 3125 for N=50000
        int waves_per_block = B / 32;                    // 8
        int pairs = (n_tiles + 1) / 2;                   // tile pairs
        int blocks = (pairs + waves_per_block - 1) / waves_per_block;
        gcn_gemm_wmma_kernel<<<blocks, B, 0, stream>>>(feat, weight, out_cnt, h, n_tiles);
    }

    // 4) edge scatter-add into output accumulator
    {
        long long total = (long long)n_edges * 16;
        int blocks = (int)((total + B - 1) / B);
        gcn_scatter_kernel<<<blocks, B, 0, stream>>>(h, src, dst, out, n_edges);
    }

    // 5) in-degree normalization + bias, in place
    {
        int total = n_nodes * OUT_FEATS;
        gcn_finalize_kernel<<<(total + B - 1) / B, B, 0, stream>>>(out, in_cnt, bias, total);
    }
}